// GATRecommender_5068061409699
// MI455X (gfx1250) — compile-verified
//
#include <hip/hip_runtime.h>
#include <math.h>

#define NEG_SLOPE 0.2f

typedef float v2f __attribute__((ext_vector_type(2)));
typedef float v8f __attribute__((ext_vector_type(8)));

__device__ __forceinline__ float lrelu(float v) { return v > 0.f ? v : NEG_SLOPE * v; }
// monotone int encoding of float for atomicMax
__device__ __forceinline__ int enc_ord(float f) {
    int i = __float_as_int(f);
    return i >= 0 ? i : (i ^ 0x7FFFFFFF);
}
__device__ __forceinline__ float dec_ord(int k) {
    return __int_as_float(k >= 0 ? k : (k ^ 0x7FFFFFFF));
}

// ---------------- init: zero accumulators, set maxes to -inf-encoding ----------------
__global__ void k_init(float* __restrict__ acc1, float* __restrict__ den1, int* __restrict__ m1,
                       float* __restrict__ acc2, float* __restrict__ den2, int* __restrict__ m2,
                       int N) {
    long i = (long)blockIdx.x * blockDim.x + threadIdx.x;
    if (i < (long)N * 128) acc1[i] = 0.f;
    if (i < (long)N * 16)  acc2[i] = 0.f;
    if (i < (long)N * 4)  { den1[i] = 0.f; m1[i] = (int)0x80000000; }
    if (i < (long)N)      { den2[i] = 0.f; m2[i] = (int)0x80000000; }
}

// ---------------- layer 1: h1 = x@W1 ; alpha_s/alpha_d per (node,head) ----------------
// block = 128 threads = one node; wave (32 lanes) = one head
__global__ void k_l1_transform(const float* __restrict__ x, const float* __restrict__ W1,
                               const float* __restrict__ asrc, const float* __restrict__ adst,
                               float* __restrict__ feat1, float* __restrict__ as1,
                               float* __restrict__ ad1) {
    int n   = blockIdx.x;
    int col = threadIdx.x;          // 0..127 = h*32 + c
    int h = col >> 5, c = col & 31;
    float x0 = x[n * 3 + 0], x1 = x[n * 3 + 1], x2 = x[n * 3 + 2];
    float hv = x0 * W1[col] + x1 * W1[128 + col] + x2 * W1[256 + col];
    feat1[(long)n * 128 + col] = hv;
    float ps = hv * asrc[col];
    float pd = hv * adst[col];
    #pragma unroll
    for (int off = 16; off > 0; off >>= 1) {
        ps += __shfl_xor(ps, off, 32);
        pd += __shfl_xor(pd, off, 32);
    }
    if (c == 0) { as1[n * 4 + h] = ps; ad1[n * 4 + h] = pd; }
}

// ---------------- layer 1: segment max over incoming edges (thread/edge) ----------------
__global__ void k_l1_edge_max(const int* __restrict__ src, const int* __restrict__ dst,
                              const float* __restrict__ as1, const float* __restrict__ ad1,
                              int* __restrict__ m1, long E, long Etot) {
    long e = (long)blockIdx.x * blockDim.x + threadIdx.x;
    if (e >= Etot) return;
    int s, d;
    if (e < E) { s = src[e]; d = dst[e]; } else { s = d = (int)(e - E); }
    #pragma unroll
    for (int h = 0; h < 4; h++) {
        float v = lrelu(as1[s * 4 + h] + ad1[d * 4 + h]);
        atomicMax(&m1[d * 4 + h], enc_ord(v));
    }
}

// ---------------- layer 1: exp, denom, weighted message scatter (wave/edge) ----------------
__global__ void k_l1_edge_acc(const int* __restrict__ src, const int* __restrict__ dst,
                              const float* __restrict__ as1, const float* __restrict__ ad1,
                              const int* __restrict__ m1, const float* __restrict__ feat1,
                              float* __restrict__ den1, float* __restrict__ acc1,
                              long E, long Etot) {
    long gid = (long)blockIdx.x * blockDim.x + threadIdx.x;
    long e   = gid >> 5;
    int lane = threadIdx.x & 31;
    if (e >= Etot) return;
    int s, d;
    if (e < E) { s = src[e]; d = dst[e]; } else { s = d = (int)(e - E); }
    float myE = 0.f;
    if (lane < 4) {
        float v = lrelu(as1[s * 4 + lane] + ad1[d * 4 + lane]);
        myE = expf(v - dec_ord(m1[d * 4 + lane]));
        atomicAdd(&den1[d * 4 + lane], myE);
    }
    const float* hs = feat1 + (long)s * 128;
    float* ao = acc1 + (long)d * 128;
    #pragma unroll
    for (int h = 0; h < 4; h++) {
        float eh = __shfl(myE, h, 32);
        atomicAdd(&ao[h * 32 + lane], eh * hs[h * 32 + lane]);
    }
}

// ---------------- layer 1 finalize: /denom + b1, ELU ; overwrite feat1 ----------------
__global__ void k_l1_final(const float* __restrict__ acc1, const float* __restrict__ den1,
                           const float* __restrict__ b1, float* __restrict__ feat1, long total) {
    long i = (long)blockIdx.x * blockDim.x + threadIdx.x;
    if (i >= total) return;
    int col = (int)(i & 127);
    long n  = i >> 7;
    int h = col >> 5;
    float dn = fmaxf(den1[n * 4 + h], 1e-16f);
    float v = acc1[i] / dn + b1[col];
    feat1[i] = v > 0.f ? v : expm1f(v);
}

// ---------------- layer 2 GEMM via WMMA: h2 = feat1 @ W2, fused alpha dot products ----
// one wave per 16-node tile; 32x V_WMMA_F32_16X16X4_F32 over K=128
__global__ void k_l2_gemm_wmma(const float* __restrict__ feat1, const float* __restrict__ W2,
                               const float* __restrict__ asrc2, const float* __restrict__ adst2,
                               float* __restrict__ h2, float* __restrict__ as2,
                               float* __restrict__ ad2, int ntiles) {
    int wave = threadIdx.x >> 5;
    int tile = blockIdx.x * 4 + wave;
    if (tile >= ntiles) return;           // wave-uniform: EXEC stays all-ones for WMMA
    int lane = threadIdx.x & 31;
    int half = lane >> 4;                 // selects K-pair (0,1) vs (2,3)
    int q    = lane & 15;                 // A: row-in-tile ; B/C: column (out channel)

    v8f acc = {0.f, 0.f, 0.f, 0.f, 0.f, 0.f, 0.f, 0.f};
    const float* arow = feat1 + (long)(tile * 16 + q) * 128;
    #pragma unroll
    for (int k0 = 0; k0 < 128; k0 += 4) {
        int k = k0 + half * 2;
        v2f a; a.x = arow[k];           a.y = arow[k + 1];
        v2f b; b.x = W2[k * 16 + q];    b.y = W2[(k + 1) * 16 + q];
        acc = __builtin_amdgcn_wmma_f32_16x16x4_f32(
            /*neg_a=*/false, a, /*neg_b=*/false, b,
            /*c_mod=*/(short)0, acc, /*reuse_a=*/false, /*reuse_b=*/false);
    }

    // C layout: lane = column (channel), VGPR r = row r (lanes 0-15) / r+8 (lanes 16-31)
    float sA = asrc2[q], sD = adst2[q];
    #pragma unroll
    for (int r = 0; r < 8; r++) {
        int m = tile * 16 + r + half * 8; // node index
        float v = acc[r];
        h2[(long)m * 16 + q] = v;
        float ps = v * sA;
        float pd = v * sD;
        #pragma unroll
        for (int off = 8; off > 0; off >>= 1) {
            ps += __shfl_xor(ps, off, 16);
            pd += __shfl_xor(pd, off, 16);
        }
        if (q == 0) { as2[m] = ps; ad2[m] = pd; }
    }
}

// ---------------- layer 2: segment max (thread/edge, H=1) ----------------
__global__ void k_l2_edge_max(const int* __restrict__ src, const int* __restrict__ dst,
                              const float* __restrict__ as2, const float* __restrict__ ad2,
                              int* __restrict__ m2, long E, long Etot) {
    long e = (long)blockIdx.x * blockDim.x + threadIdx.x;
    if (e >= Etot) return;
    int s, d;
    if (e < E) { s = src[e]; d = dst[e]; } else { s = d = (int)(e - E); }
    float v = lrelu(as2[s] + ad2[d]);
    atomicMax(&m2[d], enc_ord(v));
}

// ---------------- layer 2: exp/denom + message scatter (16 threads/edge) ----------------
__global__ void k_l2_edge_acc(const int* __restrict__ src, const int* __restrict__ dst,
                              const float* __restrict__ as2, const float* __restrict__ ad2,
                              const int* __restrict__ m2, const float* __restrict__ h2,
                              float* __restrict__ den2, float* __restrict__ acc2,
                              long E, long Etot) {
    long gid = (long)blockIdx.x * blockDim.x + threadIdx.x;
    long e = gid >> 4;
    int c  = (int)(gid & 15);
    if (e >= Etot) return;
    int s, d;
    if (e < E) { s = src[e]; d = dst[e]; } else { s = d = (int)(e - E); }
    float v  = lrelu(as2[s] + ad2[d]);
    float ev = expf(v - dec_ord(m2[d]));
    if (c == 0) atomicAdd(&den2[d], ev);
    atomicAdd(&acc2[(long)d * 16 + c], ev * h2[(long)s * 16 + c]);
}

// ---------------- layer 2 finalize: /denom + b2 -> d_out ----------------
__global__ void k_l2_final(const float* __restrict__ acc2, const float* __restrict__ den2,
                           const float* __restrict__ b2, float* __restrict__ out, long total) {
    long i = (long)blockIdx.x * blockDim.x + threadIdx.x;
    if (i >= total) return;
    int c = (int)(i & 15);
    long n = i >> 4;
    out[i] = acc2[i] / fmaxf(den2[n], 1e-16f) + b2[c];
}

extern "C" void kernel_launch(void* const* d_in, const int* in_sizes, int n_in,
                              void* d_out, int out_size, void* d_ws, size_t ws_size,
                              hipStream_t stream) {
    const float* x     = (const float*)d_in[0];
    const int*   edge  = (const int*)d_in[1];   // [2,E] int32
    const float* W1    = (const float*)d_in[2]; // [3,128]
    const float* asrc1 = (const float*)d_in[3]; // [4,32]
    const float* adst1 = (const float*)d_in[4];
    const float* b1    = (const float*)d_in[5]; // [128]
    const float* W2    = (const float*)d_in[6]; // [128,16]
    const float* asrc2 = (const float*)d_in[7]; // [1,16]
    const float* adst2 = (const float*)d_in[8];
    const float* b2    = (const float*)d_in[9]; // [16]

    const int  N    = in_sizes[0] / 3;
    const long E    = in_sizes[1] / 2;
    const long Etot = E + N;
    const int* src = edge;
    const int* dst = edge + E;

    // workspace carve-up
    float* ws    = (float*)d_ws;
    float* feat1 = ws;                       // N*128  (h1, later ELU'd features)
    float* acc1  = feat1 + (long)N * 128;    // N*128
    float* as1   = acc1  + (long)N * 128;    // N*4
    float* ad1   = as1   + (long)N * 4;      // N*4
    float* den1  = ad1   + (long)N * 4;      // N*4
    int*   m1    = (int*)(den1 + (long)N * 4);   // N*4
    float* h2    = (float*)(m1 + (long)N * 4);   // N*16
    float* acc2  = h2    + (long)N * 16;     // N*16
    float* as2   = acc2  + (long)N * 16;     // N
    float* ad2   = as2   + (long)N;          // N
    float* den2  = ad2   + (long)N;          // N
    int*   m2    = (int*)(den2 + (long)N);   // N

    const int B = 256;
    long nAll = (long)N * 128;

    k_init<<<dim3((nAll + B - 1) / B), dim3(B), 0, stream>>>(acc1, den1, m1, acc2, den2, m2, N);

    k_l1_transform<<<dim3(N), dim3(128), 0, stream>>>(x, W1, asrc1, adst1, feat1, as1, ad1);

    k_l1_edge_max<<<dim3((Etot + B - 1) / B), dim3(B), 0, stream>>>(src, dst, as1, ad1, m1, E, Etot);

    k_l1_edge_acc<<<dim3((Etot * 32 + B - 1) / B), dim3(B), 0, stream>>>(
        src, dst, as1, ad1, m1, feat1, den1, acc1, E, Etot);

    k_l1_final<<<dim3((nAll + B - 1) / B), dim3(B), 0, stream>>>(acc1, den1, b1, feat1, nAll);

    int ntiles = (N + 15) / 16;   // 3125
    k_l2_gemm_wmma<<<dim3((ntiles + 3) / 4), dim3(128), 0, stream>>>(
        feat1, W2, asrc2, adst2, h2, as2, ad2, ntiles);

    k_l2_edge_max<<<dim3((Etot + B - 1) / B), dim3(B), 0, stream>>>(src, dst, as2, ad2, m2, E, Etot);

    k_l2_edge_acc<<<dim3((Etot * 16 + B - 1) / B), dim3(B), 0, stream>>>(
        src, dst, as2, ad2, m2, h2, den2, acc2, E, Etot);

    long nOut = (long)N * 16;
    k_l2_final<<<dim3((nOut + B - 1) / B), dim3(B), 0, stream>>>(acc2, den2, b2, (float*)d_out, nOut);
}